// GNN_topexpert_69896297775690
// MI455X (gfx1250) — compile-verified
//
#include <hip/hip_runtime.h>

typedef __attribute__((ext_vector_type(2))) float v2f;
typedef __attribute__((ext_vector_type(8))) float v8f;

#define B_  4096
#define K_  32
#define D_  768
#define E_  16
#define AE_ 1024
#define GD_ 768

// ---- output offsets (floats, reference tuple order) ----
#define OFF_CLF   0ul          // [B,64,32]
#define OFF_SCA_E 8388608ul    // [B,768]
#define OFF_GRO_E 11534336ul
#define OFF_GNN   14680064ul
#define OFF_SCA_Z 17825792ul
#define OFF_GRO_Z 20971520ul
#define OFF_SCA_Q 24117248ul   // [B,16]
#define OFF_GRO_Q 24182784ul

// ---- workspace offsets (floats) ----
#define WS_SGATE 0ul
#define WS_GGATE 3145728ul
#define WS_HS    6291456ul
#define WS_HG    9437184ul
#define WS_PSUM  12582912ul   // [2][32][768]
#define WS_PSQ   12632064ul
#define WS_MU    12681216ul   // [2][768]
#define WS_RSG   12682752ul
#define WS_CNORM 12684288ul   // [32]
#define WS_MASK  12684320ul   // int

// one block: cluster squared-norms + scaffold "present" bitmask
__global__ void k_init(const int* __restrict__ nb,
                       const float* __restrict__ scac, const float* __restrict__ groc,
                       float* __restrict__ cnorm, int* __restrict__ maskp) {
  __shared__ int smask;
  int t = threadIdx.x;
  if (t == 0) smask = 0;
  __syncthreads();
  if (t < 32) {
    const float* c = (t < 16) ? (scac + (size_t)t * GD_) : (groc + (size_t)(t - 16) * GD_);
    float s = 0.f;
    for (int d = 0; d < GD_; ++d) { float v = c[d]; s += v * v; }
    cnorm[t] = s;
  }
  for (int g = t; g < B_; g += 256) {
    int mn = 0x7fffffff;
    for (int n = 0; n < K_; ++n) { int v = nb[g * K_ + n]; mn = v < mn ? v : mn; }
    mn = mn < 0 ? 0 : (mn > 7 ? 7 : mn);
    atomicOr(&smask, 1 << mn);
  }
  __syncthreads();
  if (t == 0) *maskp = smask;
}

// one block per graph: masked mean/sum pools + scaffold fixups
__global__ void k_pool(const float* __restrict__ x, const int* __restrict__ nb,
                       const int* __restrict__ sflag, const int* __restrict__ maskp,
                       float* __restrict__ out, float* __restrict__ sgate,
                       float* __restrict__ ggate) {
  __shared__ float swt[K_];
  __shared__ float rbuf[256];
  int g = blockIdx.x, t = threadIdx.x;
  if (t < K_) {
    int v = nb[g * K_ + t];
    v = v < 0 ? 0 : (v > 7 ? 7 : v);
    swt[t] = (float)((*maskp >> v) & 1);
  }
  __syncthreads();
  float gs[3] = {0.f, 0.f, 0.f}, ss[3] = {0.f, 0.f, 0.f};
  float scnt = 0.f;
  for (int n = 0; n < K_; ++n) {
    float w = swt[n];
    scnt += w;
    const float* row = x + (size_t)(g * K_ + n) * D_;
#pragma unroll
    for (int j = 0; j < 3; ++j) {
      float v = row[t + j * 256];
      gs[j] += v;
      ss[j] += w * v;
    }
  }
  float sden = scnt < 1.f ? 1.f : scnt;
  float gcnt = (float)K_ - scnt;
  float gden = gcnt < 1.f ? 1.f : gcnt;
  bool nosca = (sflag[g] == 0);
  float sca[3], gro[3], gg[3], gnn[3];
  float rsum = 0.f;
#pragma unroll
  for (int j = 0; j < 3; ++j) {
    gnn[j] = gs[j] * (1.f / (float)K_);
    sca[j] = ss[j] / sden;
    float gsum = gs[j] - ss[j];
    gro[j] = nosca ? sca[j] : (gsum / gden);
    gg[j]  = nosca ? ss[j] : gsum;
    rsum += gro[j];
  }
  rbuf[t] = rsum;
  __syncthreads();
  for (int s = 128; s > 0; s >>= 1) {
    if (t < s) rbuf[t] += rbuf[t + s];
    __syncthreads();
  }
  bool zero = (rbuf[0] == 0.f);
  size_t base = (size_t)g * D_;
#pragma unroll
  for (int j = 0; j < 3; ++j) {
    int d = t + j * 256;
    float grov = zero ? sca[j] : gro[j];
    out[OFF_GNN + base + d]   = gnn[j];
    out[OFF_SCA_E + base + d] = sca[j];
    out[OFF_GRO_E + base + d] = grov;
    sgate[base + d] = ss[j];
    ggate[base + d] = gg[j];
  }
}

// fp32 WMMA GEMM: O = A[M,KD] * W[KD,N] + bias. Block = 8 waves = 32x64 tile.
// All strides/extents are template constants so loads/stores lower to single
// base registers + immediate VGLOBAL offsets (fits signed 24-bit: <= 3.1 MB).
// MODE 0: O[r*LDO + col]; MODE 1: expert interleave into clf_logit (ldo 2048)
template <int LDA, int LDW, int LDO, int KD, int MODE, int GRP>
__global__ void k_gemm(const float* __restrict__ A,
                       const float* __restrict__ W,
                       const float* __restrict__ bias,
                       float* __restrict__ O) {
  int t = threadIdx.x;
  int lane = t & 31, wid = t >> 5;
  int wm = wid >> 2, wn = wid & 3;
  int row0 = blockIdx.y * 32 + wm * 16;
  int col0 = blockIdx.x * 64 + wn * 16;
  int m = lane & 15, hi = lane >> 4;

  float bv = bias[col0 + m];
  v8f acc;
#pragma unroll
  for (int i = 0; i < 8; ++i) acc[i] = bv;

  const float* ap = A + (size_t)(row0 + m) * LDA + 2 * hi;         // + k0
  const float* bp = W + (size_t)(2 * hi) * LDW + col0 + m;         // + k0*LDW

#pragma unroll 8
  for (int k0 = 0; k0 < KD; k0 += 4) {
    v2f a = *(const v2f*)(ap + k0);
    v2f b;
    b[0] = bp[(size_t)k0 * LDW];
    b[1] = bp[(size_t)(k0 + 1) * LDW];
    acc = __builtin_amdgcn_wmma_f32_16x16x4_f32(false, a, false, b,
                                                (short)0, acc, false, false);
  }

  int orow = row0 + hi * 8;
  if (MODE == 0) {
    float* op = O + (size_t)orow * LDO + col0 + m;
#pragma unroll
    for (int r = 0; r < 8; ++r) op[(size_t)r * LDO] = acc[r];
  } else {
    int oc = (col0 >> 4) * 32 + GRP + m;
    float* op = O + (size_t)orow * 2048 + oc;
#pragma unroll
    for (int r = 0; r < 8; ++r) op[(size_t)r * 2048] = acc[r];
  }
}

// BN stats, stage 1: per-(mat, 128-row chunk, col) partial sum / sumsq
__global__ void k_bn_part(const float* __restrict__ Hs, const float* __restrict__ Hg,
                          float* __restrict__ psum, float* __restrict__ psq) {
  int c = blockIdx.x * 256 + threadIdx.x;
  int chunk = blockIdx.y;
  int mat = blockIdx.z;
  const float* H = mat ? Hg : Hs;
  float s = 0.f, q = 0.f;
  int r0 = chunk * 128;
#pragma unroll 4
  for (int r = 0; r < 128; ++r) {
    float v = H[(size_t)(r0 + r) * GD_ + c];
    s += v; q += v * v;
  }
  size_t idx = ((size_t)mat * 32 + chunk) * GD_ + c;
  psum[idx] = s; psq[idx] = q;
}

// BN stats, stage 2: mu and rstd*gamma per column
__global__ void k_bn_final(const float* __restrict__ psum, const float* __restrict__ psq,
                           const float* __restrict__ sgam, const float* __restrict__ ggam,
                           float* __restrict__ mu, float* __restrict__ rsg) {
  int c = blockIdx.x * 256 + threadIdx.x;
  int mat = blockIdx.y;
  float s = 0.f, q = 0.f;
#pragma unroll
  for (int ch = 0; ch < 32; ++ch) {
    size_t idx = ((size_t)mat * 32 + ch) * GD_ + c;
    s += psum[idx]; q += psq[idx];
  }
  float m = s / (float)B_;
  float var = q / (float)B_ - m * m;
  float rs = rsqrtf(var + 1e-5f);
  const float* gam = mat ? ggam : sgam;
  mu[mat * GD_ + c]  = m;
  rsg[mat * GD_ + c] = rs * gam[c];
}

// BN stage 3: in-place normalize + ReLU
__global__ void k_bn_apply(float* __restrict__ Hs, float* __restrict__ Hg,
                           const float* __restrict__ mu, const float* __restrict__ rsg,
                           const float* __restrict__ sbeta, const float* __restrict__ gbeta) {
  int mat = blockIdx.y;
  float* H = mat ? Hg : Hs;
  const float* beta = mat ? gbeta : sbeta;
  size_t i = (size_t)blockIdx.x * 256 + threadIdx.x;
  int c = (int)(i % GD_);
  float v = H[i];
  v = (v - mu[mat * GD_ + c]) * rsg[mat * GD_ + c] + beta[c];
  H[i] = v > 0.f ? v : 0.f;
}

// student-t soft assignment; one block per row
__global__ void k_q(const float* __restrict__ z_all, const float* __restrict__ clus,
                    const float* __restrict__ cnorm, float* __restrict__ out_q) {
  __shared__ float sbuf[17 * 8];
  int t = threadIdx.x, lane = t & 31, wid = t >> 5;
  int b = blockIdx.x;
  const float* z = z_all + (size_t)b * GD_;
  float dot[16];
#pragma unroll
  for (int e = 0; e < 16; ++e) dot[e] = 0.f;
  float zz = 0.f;
  for (int d = t; d < GD_; d += 256) {
    float zv = z[d];
    zz += zv * zv;
#pragma unroll
    for (int e = 0; e < 16; ++e) dot[e] += zv * clus[(size_t)e * GD_ + d];
  }
#pragma unroll
  for (int v = 0; v < 17; ++v) {
    float x = (v < 16) ? dot[v] : zz;
    for (int off = 16; off; off >>= 1) x += __shfl_xor(x, off);
    if (lane == 0) sbuf[v * 8 + wid] = x;
  }
  __syncthreads();
  if (t == 0) {
    float tz = 0.f;
    for (int w = 0; w < 8; ++w) tz += sbuf[16 * 8 + w];
    float q[16], qs = 0.f;
    for (int e = 0; e < 16; ++e) {
      float td = 0.f;
      for (int w = 0; w < 8; ++w) td += sbuf[e * 8 + w];
      float d2 = tz + cnorm[e] - 2.f * td;
      d2 = d2 < 0.f ? 0.f : d2;
      q[e] = 1.f / (1.f + d2);
      qs += q[e];
    }
    for (int e = 0; e < 16; ++e) out_q[(size_t)b * 16 + e] = q[e] / qs;
  }
}

extern "C" void kernel_launch(void* const* d_in, const int* in_sizes, int n_in,
                              void* d_out, int out_size, void* d_ws, size_t ws_size,
                              hipStream_t stream) {
  (void)in_sizes; (void)n_in; (void)out_size; (void)ws_size;
  const float* node_rep = (const float*)d_in[0];
  // d_in[1] (batch) is repeat(arange(B),K): contiguity is exploited directly.
  const int* nb      = (const int*)d_in[2];
  const int* sflag   = (const int*)d_in[3];
  const float* sca_ew = (const float*)d_in[4];
  const float* sca_eb = (const float*)d_in[5];
  const float* gro_ew = (const float*)d_in[6];
  const float* gro_eb = (const float*)d_in[7];
  const float* sca_w1 = (const float*)d_in[8];
  const float* sca_b1 = (const float*)d_in[9];
  const float* sca_g  = (const float*)d_in[10];
  const float* sca_be = (const float*)d_in[11];
  const float* sca_w2 = (const float*)d_in[12];
  const float* sca_b2 = (const float*)d_in[13];
  const float* gro_w1 = (const float*)d_in[14];
  const float* gro_b1 = (const float*)d_in[15];
  const float* gro_g  = (const float*)d_in[16];
  const float* gro_be = (const float*)d_in[17];
  const float* gro_w2 = (const float*)d_in[18];
  const float* gro_b2 = (const float*)d_in[19];
  const float* sca_cl = (const float*)d_in[20];
  const float* gro_cl = (const float*)d_in[21];

  float* out = (float*)d_out;
  float* ws  = (float*)d_ws;
  int* maskp = (int*)(ws + WS_MASK);

  k_init<<<1, 256, 0, stream>>>(nb, sca_cl, gro_cl, ws + WS_CNORM, maskp);
  k_pool<<<B_, 256, 0, stream>>>(node_rep, nb, sflag, maskp, out,
                                 ws + WS_SGATE, ws + WS_GGATE);

  dim3 gE(AE_ / 64, B_ / 32);   // expert GEMMs
  dim3 gG(GD_ / 64, B_ / 32);   // gate GEMMs
  k_gemm<D_, AE_, 2048, D_, 1, 0>
      <<<gE, 256, 0, stream>>>(out + OFF_SCA_E, sca_ew, sca_eb, out + OFF_CLF);
  k_gemm<D_, AE_, 2048, D_, 1, 16>
      <<<gE, 256, 0, stream>>>(out + OFF_GRO_E, gro_ew, gro_eb, out + OFF_CLF);
  k_gemm<D_, GD_, GD_, D_, 0, 0>
      <<<gG, 256, 0, stream>>>(ws + WS_SGATE, sca_w1, sca_b1, ws + WS_HS);
  k_gemm<D_, GD_, GD_, D_, 0, 0>
      <<<gG, 256, 0, stream>>>(ws + WS_GGATE, gro_w1, gro_b1, ws + WS_HG);

  k_bn_part<<<dim3(GD_ / 256, 32, 2), 256, 0, stream>>>(ws + WS_HS, ws + WS_HG,
                                                        ws + WS_PSUM, ws + WS_PSQ);
  k_bn_final<<<dim3(GD_ / 256, 2), 256, 0, stream>>>(ws + WS_PSUM, ws + WS_PSQ,
                                                     sca_g, gro_g,
                                                     ws + WS_MU, ws + WS_RSG);
  k_bn_apply<<<dim3((B_ * GD_) / 256, 2), 256, 0, stream>>>(ws + WS_HS, ws + WS_HG,
                                                            ws + WS_MU, ws + WS_RSG,
                                                            sca_be, gro_be);

  k_gemm<GD_, GD_, GD_, GD_, 0, 0>
      <<<gG, 256, 0, stream>>>(ws + WS_HS, sca_w2, sca_b2, out + OFF_SCA_Z);
  k_gemm<GD_, GD_, GD_, GD_, 0, 0>
      <<<gG, 256, 0, stream>>>(ws + WS_HG, gro_w2, gro_b2, out + OFF_GRO_Z);

  k_q<<<B_, 256, 0, stream>>>(out + OFF_SCA_Z, sca_cl, ws + WS_CNORM,
                              out + OFF_SCA_Q);
  k_q<<<B_, 256, 0, stream>>>(out + OFF_GRO_Z, gro_cl, ws + WS_CNORM + 16,
                              out + OFF_GRO_Q);
}